// MambaModel_63385127354884
// MI455X (gfx1250) — compile-verified
//
#include <hip/hip_runtime.h>
#include <hip/hip_bf16.h>

// ---------------------------------------------------------------------------
// Mamba forward for gfx1250 (CDNA5). Matrix ops use V_WMMA_F32_16X16X4_F32
// (fp32 A/B, fp32 accumulate) so the math matches the fp32 reference exactly.
// GEMM waves are register-blocked 16x64 (4 WMMA tiles) and software-pipelined
// one K-step ahead so global loads overlap the matrix pipe.
// ---------------------------------------------------------------------------

typedef float v2f __attribute__((ext_vector_type(2)));
typedef float v8f __attribute__((ext_vector_type(8)));

#define BN 2
#define LSEQ 2048
#define DM 512
#define DI 1024
#define DSTATE 16
#define DTRANK 32
#define MROWS (BN * LSEQ)   // 4096

// ---------------------------------------------------------------------------
// h = x + pos_emb[:L]
// ---------------------------------------------------------------------------
__global__ void mamba_addpos(const float* __restrict__ x,
                             const float* __restrict__ pe,
                             float* __restrict__ h) {
  size_t i = (size_t)blockIdx.x * blockDim.x + threadIdx.x;
  size_t tot = (size_t)BN * LSEQ * DM;
  if (i >= tot) return;
  size_t ld = i % ((size_t)LSEQ * DM);
  h[i] = x[i] + pe[ld];
}

// ---------------------------------------------------------------------------
// Row LayerNorm over D=512, one 128-thread block per row.
// ---------------------------------------------------------------------------
__global__ void __launch_bounds__(128)
mamba_layernorm(const float* __restrict__ X, const float* __restrict__ g,
                const float* __restrict__ b, float* __restrict__ Y, int D) {
  __shared__ float sred[4], sred2[4];
  int row = blockIdx.x;
  const float* xr = X + (size_t)row * D;
  float s = 0.f, s2 = 0.f;
  for (int j = threadIdx.x; j < D; j += 128) {
    float v = xr[j];
    s += v; s2 += v * v;
  }
  for (int m = 16; m >= 1; m >>= 1) {
    s  += __shfl_xor(s,  m, 32);
    s2 += __shfl_xor(s2, m, 32);
  }
  int wid = threadIdx.x >> 5;
  if ((threadIdx.x & 31) == 0) { sred[wid] = s; sred2[wid] = s2; }
  __syncthreads();
  s  = sred[0] + sred[1] + sred[2] + sred[3];
  s2 = sred2[0] + sred2[1] + sred2[2] + sred2[3];
  float mu = s / (float)D;
  float var = s2 / (float)D - mu * mu;
  float rstd = rsqrtf(var + 1e-5f);
  for (int j = threadIdx.x; j < D; j += 128) {
    float v = (xr[j] - mu) * rstd;
    Y[(size_t)row * D + j] = v * g[j] + b[j];
  }
}

// ---------------------------------------------------------------------------
// WMMA fp32 GEMM: C[M,N] = act(A[M,K] @ B[K,N] + bias) + addend
// Wave tile = 16 rows x 64 cols (4 accumulators); block = 4 waves -> 64x64.
// A-frag layout (ISA 7.12.2, 32-bit A 16x4): lane half h, a[j] = A[row][k+2h+j]
// B-frag: b[j] = B[k+2h+j][col].   C: acc[r] -> row = r + 8h, col = lane&15.
// Software-pipelined: fragments for step k+1 are loaded (into distinct regs)
// before the WMMAs of step k issue, so waits are partial and loads overlap.
// act: 0 none, 1 SiLU, 2 softplus.  Requires N%64==0, M%64==0, K%4==0, K>=8.
// ---------------------------------------------------------------------------
__global__ void __launch_bounds__(128)
mamba_gemm_wmma(const float* __restrict__ A, const float* __restrict__ Bw,
                const float* __restrict__ bias, const float* addend,
                float* C, int M, int N, int K,
                int lda, int ldb, int ldc, int act) {
  int lane = threadIdx.x;                 // 0..31 (wave32)
  int wave = threadIdx.y;                 // 0..3  -> row strip within block
  int row0 = blockIdx.y * 64 + wave * 16;
  int col0 = blockIdx.x * 64;
  int half = lane >> 4, lr = lane & 15;

  // Per-lane streaming pointers (constant-stride increments in the K loop).
  const float* Ap = A  + (size_t)(row0 + lr) * lda + 2 * half;
  const float* Bp = Bw + (size_t)(2 * half) * ldb + col0 + lr;

  v8f acc[4] = {{0.f}, {0.f}, {0.f}, {0.f}};

  // ---- prologue: load fragments for K-step 0 -------------------------------
  v2f a, bf[4];
  a[0] = Ap[0];
  a[1] = Ap[1];
#pragma unroll
  for (int t = 0; t < 4; ++t) {
    bf[t][0] = Bp[t * 16];
    bf[t][1] = Bp[ldb + t * 16];
  }

  // ---- main pipelined loop: K/4 - 1 iterations -----------------------------
#pragma unroll 2
  for (int kk = 0; kk < K - 4; kk += 4) {
    Ap += 4;
    Bp += (size_t)ldb * 4;
    __builtin_prefetch(Ap + 64, 0, 3);    // A stream ~16 K-steps ahead (near)
    v2f an, bn[4];
    an[0] = Ap[0];
    an[1] = Ap[1];
#pragma unroll
    for (int t = 0; t < 4; ++t) {
      bn[t][0] = Bp[t * 16];
      bn[t][1] = Bp[ldb + t * 16];
    }
#pragma unroll
    for (int t = 0; t < 4; ++t) {
      acc[t] = __builtin_amdgcn_wmma_f32_16x16x4_f32(
          /*neg_a=*/false, a, /*neg_b=*/false, bf[t],
          /*c_mod=*/(short)0, acc[t], /*reuse_a=*/false, /*reuse_b=*/false);
    }
    a = an;
#pragma unroll
    for (int t = 0; t < 4; ++t) bf[t] = bn[t];
  }

  // ---- epilogue WMMA group (last K-step) -----------------------------------
#pragma unroll
  for (int t = 0; t < 4; ++t) {
    acc[t] = __builtin_amdgcn_wmma_f32_16x16x4_f32(
        false, a, false, bf[t], (short)0, acc[t], false, false);
  }

  // ---- store with fused bias / activation / residual -----------------------
#pragma unroll
  for (int t = 0; t < 4; ++t) {
    int col = col0 + t * 16 + lr;
    float bv = bias ? bias[col] : 0.f;
#pragma unroll
    for (int r = 0; r < 8; ++r) {
      int row = row0 + r + 8 * half;
      float v = acc[t][r] + bv;
      if (act == 1) {                     // SiLU
        v = v / (1.f + expf(-v));
      } else if (act == 2) {              // softplus
        v = (v > 20.f) ? v : log1pf(expf(v));
      }
      if (addend) v += addend[(size_t)row * ldc + col];
      C[(size_t)row * ldc + col] = v;
    }
  }
}

// ---------------------------------------------------------------------------
// Causal depthwise conv (K=4, left pad 3) + SiLU.  X,Y: [B, L, Di]
// ---------------------------------------------------------------------------
__global__ void mamba_conv_silu(const float* __restrict__ X,
                                const float* __restrict__ W,
                                const float* __restrict__ bias,
                                float* __restrict__ Y) {
  size_t idx = (size_t)blockIdx.x * blockDim.x + threadIdx.x;
  size_t tot = (size_t)BN * LSEQ * DI;
  if (idx >= tot) return;
  int d = (int)(idx % DI);
  size_t bl = idx / DI;
  int l = (int)(bl % LSEQ);
  int b = (int)(bl / LSEQ);
  const float* xb = X + (size_t)b * LSEQ * DI + d;
  float acc = bias[d];
#pragma unroll
  for (int k = 0; k < 4; ++k) {
    int ll = l - 3 + k;
    if (ll >= 0) acc += xb[(size_t)ll * DI] * W[d * 4 + k];
  }
  Y[idx] = acc / (1.f + expf(-acc));
}

// ---------------------------------------------------------------------------
// Selective scan. One lane per (b, d, n); 16-lane shuffle reduction over n.
// Writes G = (y + Dp*x) * x2  (the gated product that feeds the Wo GEMM).
// ---------------------------------------------------------------------------
__global__ void __launch_bounds__(256)
mamba_scan(const float* __restrict__ Xc, const float* __restrict__ DT,
           const float* __restrict__ Proj, const float* __restrict__ A_log,
           const float* __restrict__ Dp, const float* __restrict__ X2,
           float* __restrict__ G) {
  int gtid = blockIdx.x * 256 + threadIdx.x;
  int n  = gtid & 15;
  int ch = gtid >> 4;                     // (b, d) channel
  if (ch >= BN * DI) return;
  int d = ch % DI;
  int b = ch / DI;

  float Acoef = -expf(A_log[(size_t)d * DSTATE + n]);
  float dcoef = Dp[d];

  const float* xb  = Xc + (size_t)b * LSEQ * DI + d;
  const float* dtb = DT + (size_t)b * LSEQ * DI + d;
  const float* x2b = X2 + (size_t)b * LSEQ * DI + d;
  const float* pb  = Proj + (size_t)b * LSEQ * 64;

  float h = 0.f;
  for (int l = 0; l < LSEQ; ++l) {
    float dt = dtb[(size_t)l * DI];
    float x  = xb[(size_t)l * DI];
    float Bm = pb[(size_t)l * 64 + DTRANK + n];
    float Cm = pb[(size_t)l * 64 + DTRANK + DSTATE + n];
    h = expf(dt * Acoef) * h + dt * Bm * x;
    float y = h * Cm;
    y += __shfl_xor(y, 1, 32);
    y += __shfl_xor(y, 2, 32);
    y += __shfl_xor(y, 4, 32);
    y += __shfl_xor(y, 8, 32);
    if (n == 0) {
      G[((size_t)b * LSEQ + l) * DI + d] =
          (y + dcoef * x) * x2b[(size_t)l * DI];
    }
  }
}

// ---------------------------------------------------------------------------
extern "C" void kernel_launch(void* const* d_in, const int* in_sizes, int n_in,
                              void* d_out, int out_size, void* d_ws, size_t ws_size,
                              hipStream_t stream) {
  const float* x      = (const float*)d_in[0];
  const float* pos    = (const float*)d_in[1];
  const float* ln_g   = (const float*)d_in[2];
  const float* ln_b   = (const float*)d_in[3];
  const float* W1     = (const float*)d_in[4];
  const float* b1     = (const float*)d_in[5];
  const float* convW  = (const float*)d_in[6];
  const float* convb  = (const float*)d_in[7];
  const float* Wx     = (const float*)d_in[8];
  const float* Wdt    = (const float*)d_in[9];
  const float* bdt    = (const float*)d_in[10];
  const float* A_log  = (const float*)d_in[11];
  const float* Dp     = (const float*)d_in[12];
  const float* W2     = (const float*)d_in[13];
  const float* b2     = (const float*)d_in[14];
  const float* Wo     = (const float*)d_in[15];
  const float* bo     = (const float*)d_in[16];
  const float* fn_g   = (const float*)d_in[17];
  const float* fn_b   = (const float*)d_in[18];

  // workspace carve (floats)
  float* wsf = (float*)d_ws;
  const size_t SZ_M_DM = (size_t)MROWS * DM;   // 2,097,152
  const size_t SZ_M_DI = (size_t)MROWS * DI;   // 4,194,304
  float* bufH    = wsf;                      // [M, 512] residual stream
  float* bufXN   = bufH   + SZ_M_DM;         // [M, 512] layernormed
  float* bufX1   = bufXN  + SZ_M_DM;         // [M,1024] pre-conv / gated product
  float* bufXC   = bufX1  + SZ_M_DI;         // [M,1024] conv+silu output
  float* bufX2   = bufXC  + SZ_M_DI;         // [M,1024] gate branch
  float* bufDT   = bufX2  + SZ_M_DI;         // [M,1024] softplus(dt)
  float* bufProj = bufDT  + SZ_M_DI;         // [M,  64] x-proj (dt_in|B|C)
  (void)ws_size; (void)in_sizes; (void)n_in; (void)out_size;

  // h = x + pos_emb
  {
    size_t tot = (size_t)BN * LSEQ * DM;
    mamba_addpos<<<dim3((unsigned)((tot + 255) / 256)), dim3(256), 0, stream>>>(x, pos, bufH);
  }

  dim3 blkG(32, 4);
  for (int i = 0; i < 2; ++i) {
    // xn = LN(h)
    mamba_layernorm<<<dim3(MROWS), dim3(128), 0, stream>>>(
        bufH, ln_g + i * DM, ln_b + i * DM, bufXN, DM);

    // x1 = xn @ W1 + b1
    mamba_gemm_wmma<<<dim3(DI / 64, MROWS / 64), blkG, 0, stream>>>(
        bufXN, W1 + (size_t)i * DM * DI, b1 + i * DI, nullptr, bufX1,
        MROWS, DI, DM, DM, DI, DI, /*act=*/0);

    // xc = silu(causal_dwconv(x1))
    {
      size_t tot = (size_t)BN * LSEQ * DI;
      mamba_conv_silu<<<dim3((unsigned)((tot + 255) / 256)), dim3(256), 0, stream>>>(
          bufX1, convW + (size_t)i * DI * 4, convb + i * DI, bufXC);
    }

    // proj = xc @ Wx                      [M, 64]
    mamba_gemm_wmma<<<dim3(64 / 64, MROWS / 64), blkG, 0, stream>>>(
        bufXC, Wx + (size_t)i * DI * 64, nullptr, nullptr, bufProj,
        MROWS, 64, DI, DI, 64, 64, /*act=*/0);

    // dt = softplus(proj[:, :32] @ Wdt + bdt)   (A stride = 64)
    mamba_gemm_wmma<<<dim3(DI / 64, MROWS / 64), blkG, 0, stream>>>(
        bufProj, Wdt + (size_t)i * DTRANK * DI, bdt + i * DI, nullptr, bufDT,
        MROWS, DI, DTRANK, 64, DI, DI, /*act=*/2);

    // x2 = silu(xn @ W2 + b2)
    mamba_gemm_wmma<<<dim3(DI / 64, MROWS / 64), blkG, 0, stream>>>(
        bufXN, W2 + (size_t)i * DM * DI, b2 + i * DI, nullptr, bufX2,
        MROWS, DI, DM, DM, DI, DI, /*act=*/1);

    // selective scan -> G = (y + Dp*x) * x2  (reuse bufX1)
    mamba_scan<<<dim3((BN * DI * DSTATE) / 256), dim3(256), 0, stream>>>(
        bufXC, bufDT, bufProj, A_log + (size_t)i * DI * DSTATE,
        Dp + i * DI, bufX2, bufX1);

    // h = G @ Wo + bo + residual (in place on bufH; each element RMW by owner)
    mamba_gemm_wmma<<<dim3(DM / 64, MROWS / 64), blkG, 0, stream>>>(
        bufX1, Wo + (size_t)i * DI * DM, bo + i * DM, bufH, bufH,
        MROWS, DM, DI, DI, DM, DM, /*act=*/0);
  }

  // out = LN(h, fn_g, fn_b)
  mamba_layernorm<<<dim3(MROWS), dim3(128), 0, stream>>>(
      bufH, fn_g, fn_b, (float*)d_out, DM);
}